// SMPLModel_53171695125216
// MI455X (gfx1250) — compile-verified
//
#include <hip/hip_runtime.h>

typedef __attribute__((ext_vector_type(2))) float v2f;
typedef __attribute__((ext_vector_type(8))) float v8f;

#define NV    100000
#define NJ    24
#define NB    10
#define NP    207
#define ROWS  (NV * 3)   // 300000, divisible by 16

// workspace layout (float offsets)
#define WS_VSHAPED 0
#define WS_VPOSED  (WS_VSHAPED + ROWS)   // 300000
#define WS_J       (WS_VPOSED + ROWS)    // 600000  (24*3)
#define WS_LROT    (WS_J + NJ * 3)       // 600072  (208, padded with zero)
#define WS_REL     (WS_LROT + 208)       // 600280  (24*16)

// ---------------------------------------------------------------------------
// Kernel 1: v_shaped[r] = v_template[r] + shapedirs[r,:] . betas   (r over 3V)
// ---------------------------------------------------------------------------
__global__ void k_shape(const float* __restrict__ betas,
                        const float* __restrict__ v_template,
                        const float* __restrict__ shapedirs,
                        float* __restrict__ ws) {
  int r = blockIdx.x * blockDim.x + threadIdx.x;
  if (r >= ROWS) return;
  const float* sd = shapedirs + (size_t)r * NB;
  float acc = v_template[r];
#pragma unroll
  for (int s = 0; s < NB; ++s) acc += sd[s] * betas[s];
  ws[WS_VSHAPED + r] = acc;
}

// ---------------------------------------------------------------------------
// Kernel 2: J[j,k] = sum_v Jreg[j,v] * v_shaped[v,k]; one block per (j,k)
// ---------------------------------------------------------------------------
__global__ void k_jreg(const float* __restrict__ Jreg,
                       float* __restrict__ ws) {
  __shared__ float red[256];
  int b = blockIdx.x;          // 0..71
  int j = b / 3, k = b % 3;
  const float* row = Jreg + (size_t)j * NV;
  float acc = 0.f;
  for (int v = threadIdx.x; v < NV; v += blockDim.x)
    acc += row[v] * ws[WS_VSHAPED + v * 3 + k];
  red[threadIdx.x] = acc;
  __syncthreads();
  for (int s = 128; s > 0; s >>= 1) {
    if ((int)threadIdx.x < s) red[threadIdx.x] += red[threadIdx.x + s];
    __syncthreads();
  }
  if (threadIdx.x == 0) ws[WS_J + j * 3 + k] = red[0];
}

// ---------------------------------------------------------------------------
// Kernel 3: Rodrigues + lrotmin + forward kinematics + rel matrices (1 wave)
// ---------------------------------------------------------------------------
__global__ void k_fk(const float* __restrict__ pose, float* __restrict__ ws) {
  __shared__ float Rs[NJ][9];
  __shared__ float Gs[NJ][16];
  int lane = threadIdx.x;

  if (lane < NJ) {
    float rx = pose[lane * 3 + 0], ry = pose[lane * 3 + 1], rz = pose[lane * 3 + 2];
    float ex = rx + 1e-8f, ey = ry + 1e-8f, ez = rz + 1e-8f;
    float theta = sqrtf(ex * ex + ey * ey + ez * ez);
    float inv = 1.0f / theta;
    float ux = rx * inv, uy = ry * inv, uz = rz * inv;
    float c = cosf(theta), s = sinf(theta), oc = 1.f - c;
    float R[9];
    R[0] = c + oc * ux * ux;      R[1] = oc * ux * uy - s * uz; R[2] = oc * ux * uz + s * uy;
    R[3] = oc * uy * ux + s * uz; R[4] = c + oc * uy * uy;      R[5] = oc * uy * uz - s * ux;
    R[6] = oc * uz * ux - s * uy; R[7] = oc * uz * uy + s * ux; R[8] = c + oc * uz * uz;
#pragma unroll
    for (int i = 0; i < 9; ++i) Rs[lane][i] = R[i];
    if (lane >= 1) {
#pragma unroll
      for (int i = 0; i < 9; ++i)
        ws[WS_LROT + (lane - 1) * 9 + i] =
            R[i] - ((i == 0 || i == 4 || i == 8) ? 1.f : 0.f);
    }
  }
  if (lane == 0) ws[WS_LROT + 207] = 0.f;   // zero pad for WMMA K tail
  __syncthreads();

  if (lane == 0) {
    const int parents[NJ] = {-1, 0, 0, 0, 1, 2, 3, 4, 5, 6, 7, 8,
                             9, 9, 9, 12, 13, 14, 16, 17, 18, 19, 20, 21};
    for (int a = 0; a < 3; ++a) {
      for (int b = 0; b < 3; ++b) Gs[0][a * 4 + b] = Rs[0][a * 3 + b];
      Gs[0][a * 4 + 3] = ws[WS_J + a];
    }
    Gs[0][12] = 0.f; Gs[0][13] = 0.f; Gs[0][14] = 0.f; Gs[0][15] = 1.f;
    for (int i = 1; i < NJ; ++i) {
      int p = parents[i];
      float L[16];
      for (int a = 0; a < 3; ++a) {
        for (int b = 0; b < 3; ++b) L[a * 4 + b] = Rs[i][a * 3 + b];
        L[a * 4 + 3] = ws[WS_J + i * 3 + a] - ws[WS_J + p * 3 + a];
      }
      L[12] = 0.f; L[13] = 0.f; L[14] = 0.f; L[15] = 1.f;
      for (int a = 0; a < 4; ++a)
        for (int b = 0; b < 4; ++b) {
          float acc = 0.f;
          for (int q = 0; q < 4; ++q) acc += Gs[p][a * 4 + q] * L[q * 4 + b];
          Gs[i][a * 4 + b] = acc;
        }
    }
  }
  __syncthreads();

  if (lane < NJ) {
    float jx = ws[WS_J + lane * 3 + 0];
    float jy = ws[WS_J + lane * 3 + 1];
    float jz = ws[WS_J + lane * 3 + 2];
#pragma unroll
    for (int a = 0; a < 4; ++a) {
      float corr = Gs[lane][a * 4 + 0] * jx + Gs[lane][a * 4 + 1] * jy +
                   Gs[lane][a * 4 + 2] * jz;
      ws[WS_REL + lane * 16 + a * 4 + 0] = Gs[lane][a * 4 + 0];
      ws[WS_REL + lane * 16 + a * 4 + 1] = Gs[lane][a * 4 + 1];
      ws[WS_REL + lane * 16 + a * 4 + 2] = Gs[lane][a * 4 + 2];
      ws[WS_REL + lane * 16 + a * 4 + 3] = Gs[lane][a * 4 + 3] - corr;
    }
  }
}

// ---------------------------------------------------------------------------
// Kernel 4: v_posed = v_shaped + posedirs[3V x 207] @ lrotmin  via WMMA fp32.
// One wave per 16-row tile; K in chunks of 4; B = lrotmin broadcast over N.
// A layout: lane<16 holds A[lane][k0+0..1]; lane>=16 holds A[lane-16][k0+2..3].
// D layout: lane-half selects row-half; VGPR r = row (r | row-half*8).
// ---------------------------------------------------------------------------
__global__ void k_posedirs(const float* __restrict__ posedirs,
                           float* __restrict__ ws) {
  __shared__ float lrot[208];
  for (int i = threadIdx.x; i < 208; i += blockDim.x)
    lrot[i] = ws[WS_LROT + i];
  __syncthreads();

  int wave = threadIdx.x >> 5;
  int lane = threadIdx.x & 31;
  int tile = blockIdx.x * (blockDim.x >> 5) + wave;
  if (tile >= ROWS / 16) return;
  int base  = tile * 16;
  int m     = lane & 15;
  int khalf = (lane < 16) ? 0 : 2;
  const float* arow = posedirs + (size_t)(base + m) * NP;

  __builtin_prefetch(arow, 0, 1);            // global_prefetch_b8
  __builtin_prefetch(arow + 16 * NP, 0, 0);  // next tile, speculative

  v8f c = {};
  for (int kc = 0; kc < 52; ++kc) {
    int ka = kc * 4 + khalf;
    v2f a, b;
    a.x = (ka     < NP) ? arow[ka]     : 0.f;
    a.y = (ka + 1 < NP) ? arow[ka + 1] : 0.f;
    b.x = lrot[ka];        // index 207 reads the zero pad
    b.y = lrot[ka + 1];
    c = __builtin_amdgcn_wmma_f32_16x16x4_f32(false, a, false, b,
                                              (short)0, c, false, false);
  }
  // every D column identical (B broadcast): 16 lanes each store one row
  int rr = lane & 15;
  if (rr < 8) {
    int row = base + rr + ((lane < 16) ? 0 : 8);
    ws[WS_VPOSED + row] = c[rr] + ws[WS_VSHAPED + row];
  }
}

// ---------------------------------------------------------------------------
// Kernel 5: LBS as GEMM: out[v,a] = sum_{j,b} (w[v,j]*vh[v,b]) * rel[j,a,b]
//   = P[16 x 96] @ relFlat[96 x 16] per tile of 16 vertices (cols 0..3 used).
// ---------------------------------------------------------------------------
__global__ void k_lbs(const float* __restrict__ weights,
                      const float* __restrict__ scale,
                      const float* __restrict__ trans,
                      const float* __restrict__ ws,
                      float* __restrict__ out) {
  __shared__ float relS[NJ * 16];
  for (int i = threadIdx.x; i < NJ * 16; i += blockDim.x)
    relS[i] = ws[WS_REL + i];
  __syncthreads();

  int wave = threadIdx.x >> 5;
  int lane = threadIdx.x & 31;
  int tile = blockIdx.x * (blockDim.x >> 5) + wave;
  if (tile >= NV / 16) return;
  int base = tile * 16;
  int m = lane & 15;       // A-matrix row = vertex within tile
  int v = base + m;
  int n = lane & 15;       // B/D column
  int bhalf = (lane < 16) ? 0 : 2;

  float vh[4];
  vh[0] = ws[WS_VPOSED + v * 3 + 0];
  vh[1] = ws[WS_VPOSED + v * 3 + 1];
  vh[2] = ws[WS_VPOSED + v * 3 + 2];
  vh[3] = 1.0f;
  const float* wrow = weights + (size_t)v * NJ;

  v8f c = {};
  for (int j = 0; j < NJ; ++j) {   // K chunk j covers k = 4j .. 4j+3 (b = 0..3)
    float wj = wrow[j];
    v2f a, b;
    a.x = wj * vh[bhalf];
    a.y = wj * vh[bhalf + 1];
    // B[(j,b)][n] = rel[j][n*4 + b] for n < 4, else 0
    b.x = (n < 4) ? relS[j * 16 + n * 4 + bhalf]     : 0.f;
    b.y = (n < 4) ? relS[j * 16 + n * 4 + bhalf + 1] : 0.f;
    c = __builtin_amdgcn_wmma_f32_16x16x4_f32(false, a, false, b,
                                              (short)0, c, false, false);
  }

  int coord = lane & 15;
  if (coord < 3) {
    float sc = scale[0];
    float tr = trans[coord];
    int moff = (lane < 16) ? 0 : 8;
#pragma unroll
    for (int r = 0; r < 8; ++r)
      out[(base + moff + r) * 3 + coord] = c[r] * sc + tr;
  }
}

// ---------------------------------------------------------------------------
extern "C" void kernel_launch(void* const* d_in, const int* in_sizes, int n_in,
                              void* d_out, int out_size, void* d_ws, size_t ws_size,
                              hipStream_t stream) {
  (void)in_sizes; (void)n_in; (void)out_size; (void)ws_size;
  const float* betas      = (const float*)d_in[0];
  const float* pose       = (const float*)d_in[1];
  const float* scale      = (const float*)d_in[2];
  const float* trans      = (const float*)d_in[3];
  const float* v_template = (const float*)d_in[4];
  const float* shapedirs  = (const float*)d_in[5];
  const float* posedirs   = (const float*)d_in[6];
  const float* Jreg       = (const float*)d_in[7];
  const float* weights    = (const float*)d_in[8];
  float* out = (float*)d_out;
  float* ws  = (float*)d_ws;

  k_shape<<<(ROWS + 255) / 256, 256, 0, stream>>>(betas, v_template, shapedirs, ws);
  k_jreg<<<NJ * 3, 256, 0, stream>>>(Jreg, ws);
  k_fk<<<1, 32, 0, stream>>>(pose, ws);
  k_posedirs<<<(ROWS / 16 + 7) / 8, 256, 0, stream>>>(posedirs, ws);
  k_lbs<<<(NV / 16 + 7) / 8, 256, 0, stream>>>(weights, scale, trans, ws, out);
}